// UpsampleFlow_9354438770960
// MI455X (gfx1250) — compile-verified
//
#include <hip/hip_runtime.h>
#include <hip/hip_bf16.h>

typedef float v2f __attribute__((ext_vector_type(2)));
typedef float v8f __attribute__((ext_vector_type(8)));

#define B_  8
#define N_  8192
#define S_  2048
#define EPS_ 1e-10f

// One workgroup = 8 wave32s, each wave owns a 16-wide tile of dense points n.
//
// LDS:
//   ldsA[s]    = (-2*sx, -2*sy, -2*sz, |s|^2)  -- WMMA A-fragment table (32 KB)
//   ldsFlow    = sparse_flow[b] staged via global_load_async_to_lds_b128 (24 KB),
//                copy overlapped with the whole distance loop (ASYNCcnt).
//
// Ranking key(s,n) = -2 s.x + |s|^2 : ordering over s identical to d^2 since
// |x|^2 is a per-column constant. One V_WMMA_F32_16X16X4_F32 (K=4 >= 3, exact
// f32) produces a 16(s) x 16(n) key tile; each lane keeps a lane-local top-3
// via an exact paired-swap network built from v_cmp + v_cndmask (one compare
// drives both the key and index selects -> no fminnum canonicalize overhead).
__global__ __launch_bounds__(256)
void upsample_flow_3nn_kernel(const float* __restrict__ xyz,         // [B,3,N]
                              const float* __restrict__ sparse_xyz,  // [B,3,S]
                              const float* __restrict__ sparse_flow, // [B,3,S]
                              float* __restrict__ out)               // [B,3,N]
{
    __shared__ float4 ldsA[S_];                         // 32 KB
    __shared__ __align__(16) float ldsFlow[3 * S_];     // 24 KB

    const int tid    = threadIdx.x;
    const int wave   = tid >> 5;
    const int lane   = tid & 31;
    const int laneLo = lane & 15;
    const bool hi    = (lane >= 16);

    const int wgPerB = N_ / 128;                 // 64 workgroups per batch
    const int b      = blockIdx.x / wgPerB;
    const int nBase  = (blockIdx.x % wgPerB) * 128 + wave * 16;

    // lane-local top-3 (ascending keys), exact f32 + index
    float k0 = 3.0e38f, k1 = 3.0e38f, k2 = 3.0e38f;
    int   i0 = 0,       i1 = 0,       i2 = 0;

    // exact, stable sorted-insert of (key, si) into the top-3
    auto insert3 = [&](float key, int si) {
        const bool  c0 = key < k0;
        const float kx = c0 ? k0 : key;
        const int   ix = c0 ? i0 : si;
        k0 = c0 ? key : k0;
        i0 = c0 ? si  : i0;
        const bool  c1 = kx < k1;
        const float ky = c1 ? k1 : kx;
        const int   iy = c1 ? i1 : ix;
        k1 = c1 ? kx : k1;
        i1 = c1 ? ix : i1;
        const bool  c2 = ky < k2;
        k2 = c2 ? ky : k2;
        i2 = c2 ? iy : i2;
    };

    // ---- Kick off async copy: sparse_flow[b] (6144 floats = 1536 x b128) ----
    // Issued before everything else; overlaps table staging + distance loop.
    {
        const unsigned long long fbase =
            (unsigned long long)(sparse_flow + (size_t)b * 3 * S_);
        const unsigned ldsBase = (unsigned)(size_t)(&ldsFlow[0]); // LDS byte off
#pragma unroll
        for (int k = 0; k < 6; ++k) {
            const unsigned byteOff = (unsigned)(tid + k * 256) * 16u;
            const unsigned dstAddr = ldsBase + byteOff;
            asm volatile("global_load_async_to_lds_b128 %0, %1, %2 offset:0"
                         :
                         : "v"(dstAddr), "v"(byteOff), "s"(fbase)
                         : "memory");
        }
    }

    // ---- Stage transformed sparse coords into LDS (coalesced) ----
    const float* sp = sparse_xyz + (size_t)b * 3 * S_;
    for (int s = tid; s < S_; s += 256) {
        const float px = sp[0 * S_ + s];
        const float py = sp[1 * S_ + s];
        const float pz = sp[2 * S_ + s];
        float4 v;
        v.x = -2.0f * px;
        v.y = -2.0f * py;
        v.z = -2.0f * pz;
        v.w = px * px + py * py + pz * pz;
        ldsA[s] = v;
    }
    __syncthreads();

    // ---- B fragment (constant over the whole s loop) ----
    // B is K(4) x N(16): k0=xx, k1=xy, k2=xz, k3=1. Lane owns column n.
    const float* xp = xyz + (size_t)b * 3 * N_;
    const int n = nBase + laneLo;
    v2f bfrag;
    bfrag.x = hi ? xp[2 * N_ + n] : xp[0 * N_ + n];
    bfrag.y = hi ? 1.0f           : xp[1 * N_ + n];

    // ---- Main loop: 128 chunks, software-pipelined A-fragment loads ----
    const float* ldsAf  = (const float*)&ldsA[0];
    const int laneOffA  = (laneLo << 2) + (hi ? 2 : 0);  // float index in table
    v2f aCur;
    aCur.x = ldsAf[laneOffA];
    aCur.y = ldsAf[laneOffA + 1];

    for (int sb = 0; sb < S_; sb += 16) {
        const v2f a = aCur;

        // prefetch next chunk's fragment (wraps to 0 on last iter; harmless)
        const int sbN = (sb + 16) & (S_ - 1);
        aCur.x = ldsAf[sbN * 4 + laneOffA];
        aCur.y = ldsAf[sbN * 4 + laneOffA + 1];

        v8f c = {};
        v8f d = __builtin_amdgcn_wmma_f32_16x16x4_f32(
            /*neg_a=*/false, a, /*neg_b=*/false, bfrag,
            /*c_mod=*/(short)0, c, /*reuse_a=*/false, /*reuse_b=*/false);

        // D rows: lane<16 holds s_local = r, lane>=16 holds s_local = 8+r
        const int sloc = sb + (hi ? 8 : 0);
#pragma unroll
        for (int r = 0; r < 8; ++r) {
            insert3(d[r], sloc + r);
        }
    }

    // ---- Merge the two half-wave partial top-3 lists (lane ^ 16) ----
    {
        const float pk[3] = {__shfl_xor(k0, 16, 32), __shfl_xor(k1, 16, 32),
                             __shfl_xor(k2, 16, 32)};
        const int   pi[3] = {__shfl_xor(i0, 16, 32), __shfl_xor(i1, 16, 32),
                             __shfl_xor(i2, 16, 32)};
#pragma unroll
        for (int j = 0; j < 3; ++j) {
            insert3(pk[j], pi[j]);
        }
    }

    // ---- Wait for async flow copy, then all-wave barrier ----
    asm volatile("s_wait_asynccnt 0x0" ::: "memory");
    __syncthreads();

    // ---- Epilogue: lanes 0..15 finalize one dense point each ----
    if (!hi) {
        const float x0 = xp[0 * N_ + n];
        const float x1 = xp[1 * N_ + n];
        const float x2 = xp[2 * N_ + n];

        const int idx[3] = {i0, i1, i2};
        float w[3];
        float wsum = 0.0f;
#pragma unroll
        for (int j = 0; j < 3; ++j) {
            const float* a = (const float*)&ldsA[idx[j]];
            const float sx = -0.5f * a[0];
            const float sy = -0.5f * a[1];
            const float sz = -0.5f * a[2];
            const float dx = x0 - sx;
            const float dy = x1 - sy;
            const float dz = x2 - sz;
            float dist = sqrtf(dx * dx + dy * dy + dz * dz);
            dist = fmaxf(dist, EPS_);
            w[j] = 1.0f / dist;
            wsum += w[j];
        }
        const float invw = 1.0f / wsum;

        float o0 = 0.0f, o1 = 0.0f, o2 = 0.0f;
#pragma unroll
        for (int j = 0; j < 3; ++j) {
            const float wj = w[j] * invw;
            const int s = idx[j];
            o0 += wj * ldsFlow[0 * S_ + s];
            o1 += wj * ldsFlow[1 * S_ + s];
            o2 += wj * ldsFlow[2 * S_ + s];
        }
        o0 = fminf(fmaxf(o0, -100.0f), 100.0f);
        o1 = fminf(fmaxf(o1, -100.0f), 100.0f);
        o2 = fminf(fmaxf(o2, -100.0f), 100.0f);

        float* op = out + (size_t)b * 3 * N_;
        op[0 * N_ + n] = o0;
        op[1 * N_ + n] = o1;
        op[2 * N_ + n] = o2;
    }
}

extern "C" void kernel_launch(void* const* d_in, const int* in_sizes, int n_in,
                              void* d_out, int out_size, void* d_ws, size_t ws_size,
                              hipStream_t stream) {
    (void)in_sizes; (void)n_in; (void)out_size; (void)d_ws; (void)ws_size;
    const float* xyz         = (const float*)d_in[0];
    const float* sparse_xyz  = (const float*)d_in[1];
    const float* sparse_flow = (const float*)d_in[2];
    float* out               = (float*)d_out;

    const dim3 grid(B_ * (N_ / 128));   // 512 workgroups
    const dim3 block(256);              // 8 wave32s
    upsample_flow_3nn_kernel<<<grid, block, 0, stream>>>(xyz, sparse_xyz,
                                                         sparse_flow, out);
}